// GRF_HGNN_K4_22514218565925
// MI455X (gfx1250) — compile-verified
//
#include <hip/hip_runtime.h>
#include <hip/hip_bf16.h>

// ---------------------------------------------------------------------------
// GRF hetero-GNN forward for MI455X (gfx1250), fp32 via V_WMMA_F32_16X16X4_F32.
//   - gemm128: fully-unrolled K=128 wave32 WMMA tile engine. Full unroll folds
//     all k*4 terms into load immediate-offset fields (zero in-loop address
//     math) and lets the scheduler hoist load clauses ahead of their WMMAs.
//     Workgroup = 4 waves, tile = 64 rows x 128 cols, wave = 4 Mt x 2 Nt.
//   - encoder GEMM: branchless generic-K variant, per-row symmetry coefficient
//     hoisted out of the K loop, clamped+masked tail (EXEC stays all-ones).
//   - scatter: wave per edge, scalarized index loads, f32 atomics (L2-hot).
// ---------------------------------------------------------------------------

typedef float v2f __attribute__((ext_vector_type(2)));
typedef float v4f __attribute__((ext_vector_type(4)));
typedef float v8f __attribute__((ext_vector_type(8)));

#define HID 128

__device__ __forceinline__ v8f wmma_f32x4(v2f a, v2f b, v8f c) {
  // D = A(16x4 f32) * B(4x16 f32) + C(16x16 f32)
  return __builtin_amdgcn_wmma_f32_16x16x4_f32(
      /*neg_a=*/false, a, /*neg_b=*/false, b,
      /*c_mod=*/(short)0, c, /*reuse_a=*/false, /*reuse_b=*/false);
}

// ---------------------------------------------------------------------------
// Fixed-K (K=128) GEMM: C[N x 128] (+)= A[N x 128] @ W[128 x 128] (+bias)(relu)
// ---------------------------------------------------------------------------
template <int ACC, int RELU, int BIAS>
__global__ __launch_bounds__(128) void gemm128_kernel(
    const float* __restrict__ A, const float* __restrict__ W,
    const float* __restrict__ bias, float* __restrict__ C, int N) {
  const int wave = threadIdx.x >> 5;
  const int lane = threadIdx.x & 31;
  const int lm   = lane & 15;
  const int hi   = lane >> 4;          // 0: lanes 0-15, 1: lanes 16-31
  const int m0   = blockIdx.x * 64;    // 64 rows per workgroup
  const int n0   = wave * 32;          // 32 cols per wave
  if (m0 >= N) return;                 // wave-uniform

  // Per-lane base pointers; all k-dependent addressing becomes an immediate.
  const float* Arow[4];
#pragma unroll
  for (int mt = 0; mt < 4; ++mt)
    Arow[mt] = A + (size_t)(m0 + mt * 16 + lm) * 128 + 2 * hi;
  const float* Wb0 = W + (size_t)(2 * hi) * 128 + n0 + lm;      // row k+2*hi
  const float* Wb1 = W + (size_t)(2 * hi + 1) * 128 + n0 + lm;  // row k+2*hi+1

  v8f acc[4][2] = {};

#pragma unroll
  for (int k = 0; k < 128; k += 4) {
    // A fragment: 16x4 f32, lane holds A[row][k+2*hi .. k+2*hi+1]
    v2f a[4];
#pragma unroll
    for (int mt = 0; mt < 4; ++mt) a[mt] = *(const v2f*)(Arow[mt] + k);
    // B fragment: 4x16 f32, v0 = row K=k(+2), v1 = row K=k+1(+3)
    v2f b[2];
#pragma unroll
    for (int j = 0; j < 2; ++j) {
      b[j].x = Wb0[(size_t)k * 128 + j * 16];
      b[j].y = Wb1[(size_t)k * 128 + j * 16];
    }
#pragma unroll
    for (int mt = 0; mt < 4; ++mt)
#pragma unroll
      for (int j = 0; j < 2; ++j)
        acc[mt][j] = wmma_f32x4(a[mt], b[j], acc[mt][j]);
  }

  // C/D layout: VGPR r, lanes 0-15 -> M=r, lanes 16-31 -> M=r+8, N=lane&15
#pragma unroll
  for (int mt = 0; mt < 4; ++mt)
#pragma unroll
    for (int j = 0; j < 2; ++j)
#pragma unroll
      for (int r = 0; r < 8; ++r) {
        const int row = m0 + mt * 16 + hi * 8 + r;
        const int col = n0 + j * 16 + lm;
        float v = acc[mt][j][r];
        if (BIAS) v += bias[col];
        if (RELU) v = v > 0.f ? v : 0.f;
        float* p = C + (size_t)row * 128 + col;
        if (ACC) v += *p;
        *p = v;
      }
}

// ---------------------------------------------------------------------------
// Generic-K encoder GEMM: C[N x 128] = relu(A[N x K] @ W[K x 128] + bias).
// HAS_COEF: per-row scale coeff[row % 12] hoisted out of the K loop.
// Main loop over K&~3 is unguarded; tail uses clamped indices + 0/1 masks
// (branchless; EXEC remains all-ones).
// ---------------------------------------------------------------------------
template <int HAS_COEF>
__global__ __launch_bounds__(128) void gemm_encode_kernel(
    const float* __restrict__ A, const float* __restrict__ W,
    const float* __restrict__ bias, const float* __restrict__ rowcoef,
    float* __restrict__ C, int N, int K) {
  const int wave = threadIdx.x >> 5;
  const int lane = threadIdx.x & 31;
  const int lm   = lane & 15;
  const int hi   = lane >> 4;
  const int m0   = blockIdx.x * 64;
  const int n0   = wave * 32;
  if (m0 >= N) return;

  // Hoisted per-row scales (joint symmetry coefficients).
  float s[4];
#pragma unroll
  for (int mt = 0; mt < 4; ++mt)
    s[mt] = HAS_COEF ? rowcoef[(m0 + mt * 16 + lm) % 12] : 1.0f;

  const float* Arow[4];
#pragma unroll
  for (int mt = 0; mt < 4; ++mt)
    Arow[mt] = A + (size_t)(m0 + mt * 16 + lm) * K;

  v8f acc[4][2] = {};
  const int K4 = K & ~3;

#pragma unroll 2
  for (int k = 0; k < K4; k += 4) {
    const int k0 = k + 2 * hi;  // k0+1 <= K4-1 < K: unguarded
    v2f a[4];
#pragma unroll
    for (int mt = 0; mt < 4; ++mt) {
      v2f t = *(const v2f*)(Arow[mt] + k0);
      a[mt].x = t.x * s[mt];
      a[mt].y = t.y * s[mt];
    }
    v2f b[2];
#pragma unroll
    for (int j = 0; j < 2; ++j) {
      const int n = n0 + j * 16 + lm;
      b[j].x = W[(size_t)k0 * 128 + n];
      b[j].y = W[(size_t)(k0 + 1) * 128 + n];
    }
#pragma unroll
    for (int mt = 0; mt < 4; ++mt)
#pragma unroll
      for (int j = 0; j < 2; ++j)
        acc[mt][j] = wmma_f32x4(a[mt], b[j], acc[mt][j]);
  }

  if (K4 < K) {  // branchless masked tail (uniform condition)
    const int k0 = K4 + 2 * hi;
    const float m0f = (k0 < K) ? 1.f : 0.f;
    const float m1f = (k0 + 1 < K) ? 1.f : 0.f;
    const int i0 = (k0 < K) ? k0 : 0;          // clamped safe index
    const int i1 = (k0 + 1 < K) ? k0 + 1 : 0;
    v2f a[4];
#pragma unroll
    for (int mt = 0; mt < 4; ++mt) {
      a[mt].x = Arow[mt][i0] * (s[mt] * m0f);
      a[mt].y = Arow[mt][i1] * (s[mt] * m1f);
    }
    v2f b[2];
#pragma unroll
    for (int j = 0; j < 2; ++j) {
      const int n = n0 + j * 16 + lm;
      b[j].x = W[(size_t)i0 * 128 + n] * m0f;
      b[j].y = W[(size_t)i1 * 128 + n] * m1f;
    }
#pragma unroll
    for (int mt = 0; mt < 4; ++mt)
#pragma unroll
      for (int j = 0; j < 2; ++j)
        acc[mt][j] = wmma_f32x4(a[mt], b[j], acc[mt][j]);
  }

#pragma unroll
  for (int mt = 0; mt < 4; ++mt)
#pragma unroll
    for (int j = 0; j < 2; ++j)
#pragma unroll
      for (int r = 0; r < 8; ++r) {
        const int row = m0 + mt * 16 + hi * 8 + r;
        const int col = n0 + j * 16 + lm;
        float v = acc[mt][j][r] + bias[col];
        C[(size_t)row * 128 + col] = v > 0.f ? v : 0.f;
      }
}

// ---------------------------------------------------------------------------
// Edge scatter-aggregate: one wave per edge; each lane moves 4 features.
// Edge index forced scalar so src/dst indices load through SMEM.
// ---------------------------------------------------------------------------
__global__ void scatter_add_kernel(const int* __restrict__ src,
                                   const int* __restrict__ dst,
                                   const float* __restrict__ X,
                                   float* __restrict__ AGG,
                                   float* __restrict__ CNT, int E) {
  const int gid  = blockIdx.x * blockDim.x + threadIdx.x;
  const int e    = __builtin_amdgcn_readfirstlane(gid >> 5);
  const int lane = threadIdx.x & 31;
  if (e >= E) return;
  const int s = src[e];
  const int d = dst[e];
  v4f v = *(const v4f*)(X + (size_t)s * 128 + lane * 4);
  float* out = AGG + (size_t)d * 128 + lane * 4;
  atomicAdd(out + 0, v.x);
  atomicAdd(out + 1, v.y);
  atomicAdd(out + 2, v.z);
  atomicAdd(out + 3, v.w);
  if (CNT != nullptr && lane == 0) atomicAdd(CNT + d, 1.0f);
}

// AGG[row][*] /= max(CNT[row],1); processed as float4 (row = 4*i4 >> 7)
__global__ void mean_div_kernel(v4f* __restrict__ AGG,
                                const float* __restrict__ CNT, size_t n4) {
  size_t i = (size_t)blockIdx.x * blockDim.x + threadIdx.x;
  if (i >= n4) return;
  float c = CNT[(i * 4) >> 7];
  c = c > 1.f ? c : 1.f;
  const float r = 1.f / c;
  v4f v = AGG[i];
  v.x *= r; v.y *= r; v.z *= r; v.w *= r;
  AGG[i] = v;
}

__global__ void fill_zero_kernel(v4f* __restrict__ p, size_t n4) {
  size_t i = (size_t)blockIdx.x * blockDim.x + threadIdx.x;
  const size_t stride = (size_t)gridDim.x * blockDim.x;
  const v4f z = {0.f, 0.f, 0.f, 0.f};
  for (; i < n4; i += stride) p[i] = z;
}

// x += (relu? max(nw,0) : nw)   -- residual update, float4
__global__ void residual_kernel(v4f* __restrict__ x, const v4f* __restrict__ nw,
                                size_t n4, int relu) {
  size_t i = (size_t)blockIdx.x * blockDim.x + threadIdx.x;
  const size_t stride = (size_t)gridDim.x * blockDim.x;
  for (; i < n4; i += stride) {
    v4f v = nw[i];
    if (relu) {
      v.x = v.x > 0.f ? v.x : 0.f;
      v.y = v.y > 0.f ? v.y : 0.f;
      v.z = v.z > 0.f ? v.z : 0.f;
      v.w = v.w > 0.f ? v.w : 0.f;
    }
    v4f o = x[i];
    o.x += v.x; o.y += v.y; o.z += v.z; o.w += v.w;
    x[i] = o;
  }
}

// out[row] = dot(Xf[row], Wd) + bd : wave-per-row, wave32 shfl reduction
__global__ void decode_kernel(const float* __restrict__ Xf,
                              const float* __restrict__ Wd,
                              const float* __restrict__ bd,
                              float* __restrict__ out, int N) {
  const int gid  = blockIdx.x * blockDim.x + threadIdx.x;
  const int row  = gid >> 5;
  const int lane = gid & 31;
  if (row >= N) return;
  v4f x = *(const v4f*)(Xf + (size_t)row * 128 + lane * 4);
  v4f w = *(const v4f*)(Wd + lane * 4);
  float p = x.x * w.x + x.y * w.y + x.z * w.z + x.w * w.w;
  for (int off = 16; off > 0; off >>= 1) p += __shfl_xor(p, off, 32);
  if (lane == 0) out[row] = p + bd[0];
}

// ---------------------------------------------------------------------------
// Host orchestration.
// Input flattening (setup_inputs dict insertion order, depth-first):
//   0:x_base 1:x_joint 2:x_foot 3:joint_coeffs
//   4..9: enc (base W,b | joint W,b | foot W,b)
//   10 + l*24 + e*3 + {0:W_rel, 1:b_rel, 2:W_root}, e in
//        [gt, gs, gr, bj, jb, jj, jf, fj], l in 0..3
//   106:W1 107:b1 108:W2 109:b2  110:Wd 111:bd
//   112..119: edges [2,E] (src then dst), EDGE_DEFS order
// ---------------------------------------------------------------------------
extern "C" void kernel_launch(void* const* d_in, const int* in_sizes, int n_in,
                              void* d_out, int out_size, void* d_ws,
                              size_t ws_size, hipStream_t stream) {
  (void)in_sizes; (void)out_size; (void)ws_size;
  if (n_in < 120) return;

  const int N0 = 32768, N1 = 98304, N2 = 32768;  // base, joint, foot

  const float* x_base  = (const float*)d_in[0];
  const float* x_joint = (const float*)d_in[1];
  const float* x_foot  = (const float*)d_in[2];
  const float* coeffs  = (const float*)d_in[3];
  const float* encW[3] = {(const float*)d_in[4], (const float*)d_in[6],
                          (const float*)d_in[8]};
  const float* encB[3] = {(const float*)d_in[5], (const float*)d_in[7],
                          (const float*)d_in[9]};
  const float* W1 = (const float*)d_in[106];
  const float* b1 = (const float*)d_in[107];
  const float* W2 = (const float*)d_in[108];
  const float* b2 = (const float*)d_in[109];
  const float* Wd = (const float*)d_in[110];
  const float* bd = (const float*)d_in[111];

  // Workspace layout
  float* ws = (float*)d_ws;
  size_t off = 0;
  auto alloc = [&](size_t n) { float* p = ws + off; off += n; return p; };
  float* xb  = alloc((size_t)N0 * 128);
  float* xj  = alloc((size_t)N1 * 128);
  float* xf  = alloc((size_t)N2 * 128);
  float* nb  = alloc((size_t)N0 * 128);
  float* nj  = alloc((size_t)N1 * 128);
  float* nf  = alloc((size_t)N2 * 128);
  float* agg = alloc((size_t)N1 * 128);  // sized for largest target (joint)
  float* cnt = alloc((size_t)N0);        // mean types target base only
  float* tmp = alloc((size_t)N0 * 128);  // base-MLP hidden

  // Encoders (joint symmetry coeff folded into A load: coeff[row % 12])
  gemm_encode_kernel<0><<<N0 / 64, 128, 0, stream>>>(x_base, encW[0], encB[0],
                                                     nullptr, xb, N0, 6);
  gemm_encode_kernel<1><<<N1 / 64, 128, 0, stream>>>(x_joint, encW[1], encB[1],
                                                     coeffs, xj, N1, 300);
  gemm_encode_kernel<0><<<N2 / 64, 128, 0, stream>>>(x_foot, encW[2], encB[2],
                                                     nullptr, xf, N2, 4);

  // Edge metadata (EDGE_DEFS order): src type, dst type, mean?, #edges
  struct EdgeMeta { int s, t, mean, E; };
  const EdgeMeta em[8] = {
      {0, 0, 1, 32768},   // base  gt base  mean
      {0, 0, 1, 32768},   // base  gs base  mean
      {0, 0, 0, 32768},   // base  gr base  add
      {0, 1, 0, 98304},   // base  bj joint add
      {1, 0, 0, 98304},   // joint jb base  add
      {1, 1, 0, 131072},  // joint jj joint add
      {1, 2, 0, 32768},   // joint jf foot  add
      {2, 1, 0, 32768},   // foot  fj joint add
  };
  float* X[3]  = {xb, xj, xf};
  float* NW[3] = {nb, nj, nf};
  const int NN[3] = {N0, N1, N2};

  for (int l = 0; l < 4; ++l) {
    bool first[3] = {true, true, true};
    for (int e = 0; e < 8; ++e) {
      const int s = em[e].s, t = em[e].t, E = em[e].E, Nt = NN[t];
      const int* ei  = (const int*)d_in[112 + e];
      const int* src = ei;
      const int* dst = ei + E;
      const size_t nagg4 = (size_t)Nt * 32;  // float4 count

      fill_zero_kernel<<<2048, 256, 0, stream>>>((v4f*)agg, nagg4);
      if (em[e].mean)
        fill_zero_kernel<<<64, 256, 0, stream>>>((v4f*)cnt, (size_t)Nt / 4);

      scatter_add_kernel<<<(E * 32 + 255) / 256, 256, 0, stream>>>(
          src, dst, X[s], agg, em[e].mean ? cnt : nullptr, E);
      if (em[e].mean)
        mean_div_kernel<<<(int)((nagg4 + 255) / 256), 256, 0, stream>>>(
            (v4f*)agg, cnt, nagg4);

      const float* Wrel  = (const float*)d_in[10 + l * 24 + e * 3 + 0];
      const float* brel  = (const float*)d_in[10 + l * 24 + e * 3 + 1];
      const float* Wroot = (const float*)d_in[10 + l * 24 + e * 3 + 2];

      // new[t] (+)= agg @ W_rel + b_rel
      if (first[t]) {
        gemm128_kernel<0, 0, 1><<<Nt / 64, 128, 0, stream>>>(agg, Wrel, brel,
                                                             NW[t], Nt);
        first[t] = false;
      } else {
        gemm128_kernel<1, 0, 1><<<Nt / 64, 128, 0, stream>>>(agg, Wrel, brel,
                                                             NW[t], Nt);
      }
      // new[t] += x[t] @ W_root
      gemm128_kernel<1, 0, 0><<<Nt / 64, 128, 0, stream>>>(X[t], Wroot, nullptr,
                                                           NW[t], Nt);
    }

    // base MLP: nb = relu(nb@W1 + b1) @ W2 + b2
    gemm128_kernel<0, 1, 1><<<N0 / 64, 128, 0, stream>>>(nb, W1, b1, tmp, N0);
    gemm128_kernel<0, 0, 1><<<N0 / 64, 128, 0, stream>>>(tmp, W2, b2, nb, N0);

    // residuals: x = new + x (joint/foot with relu(new))
    residual_kernel<<<2048, 256, 0, stream>>>((v4f*)xb, (const v4f*)nb,
                                              (size_t)N0 * 32, 0);
    residual_kernel<<<4096, 256, 0, stream>>>((v4f*)xj, (const v4f*)nj,
                                              (size_t)N1 * 32, 1);
    residual_kernel<<<2048, 256, 0, stream>>>((v4f*)xf, (const v4f*)nf,
                                              (size_t)N2 * 32, 1);
  }

  // Decoder: out = x_foot @ Wd + bd
  decode_kernel<<<(N2 * 32 + 255) / 256, 256, 0, stream>>>(xf, Wd, bd,
                                                           (float*)d_out, N2);
}